// GraphConvolutionK_61203283968718
// MI455X (gfx1250) — compile-verified
//
#include <hip/hip_runtime.h>

#define K_SLICES 8
#define DIN 128
#define DOUT 128

typedef __attribute__((ext_vector_type(2))) float v2f;
typedef __attribute__((ext_vector_type(8))) float v8f;

// ---------------------------------------------------------------- degrees ---
__global__ void deg_init_kernel(float* __restrict__ deg, int nN) {
  int i = blockIdx.x * blockDim.x + threadIdx.x;
  if (i < nN) deg[i] = 1.0f;  // self-loop contribution
}

__global__ void deg_count_kernel(const int* __restrict__ dst,
                                 float* __restrict__ deg, int nE) {
  int e = blockIdx.x * blockDim.x + threadIdx.x;
  if (e < nE) unsafeAtomicAdd(&deg[dst[e]], 1.0f);
}

__global__ void deg_rsqrt_kernel(float* __restrict__ deg, int nN) {
  int i = blockIdx.x * blockDim.x + threadIdx.x;
  if (i < nN) deg[i] = rsqrtf(deg[i]);  // deg >= 1 always
}

// ------------------------------------------------------- H = x[:,k,:] @ W ---
// One wave32 computes one 16x16 output tile with V_WMMA_F32_16X16X4_F32,
// stepping K in chunks of 4 (32 WMMA ops per tile).
// A (16x4 f32, 2 VGPRs): lanes 0-15 hold M=lane, K = kb+0/+1;
//                        lanes 16-31 hold M=lane-16, K = kb+2/+3.
// B (4x16 f32, 2 VGPRs): mirrored (lanes give N, vgpr gives K).
// C/D (16x16 f32, 8 VGPRs): vgpr r -> M = r (+8 for lanes 16-31), N = lane&15.
__global__ __launch_bounds__(128) void gemm_wmma_kernel(
    const float* __restrict__ x,   // [N, K_SLICES, DIN]
    const float* __restrict__ Wm,  // [DIN, DOUT]
    float* __restrict__ H,         // [N, DOUT]
    int kslice, int nN)
{
  const int lane  = threadIdx.x & 31;
  const int wave  = blockIdx.x * (blockDim.x >> 5) + (threadIdx.x >> 5);
  const int tilesN = DOUT / 16;                 // 8
  const int tileM  = wave / tilesN;
  const int tileN  = wave % tilesN;
  if (tileM * 16 >= nN) return;                 // wave-uniform: EXEC stays all-1s

  int row = tileM * 16 + (lane & 15);
  if (row >= nN) row = nN - 1;                  // tail clamp (N%16==0 here)
  const int col  = tileN * 16 + (lane & 15);
  const int koff = (lane >> 4) << 1;            // 0 (lanes 0-15) or 2 (lanes 16-31)

  const float* xr = x + ((size_t)row * K_SLICES + kslice) * DIN;

  v8f c = {0.f, 0.f, 0.f, 0.f, 0.f, 0.f, 0.f, 0.f};
#pragma unroll
  for (int kb = 0; kb < DIN; kb += 4) {
    v2f a, b;
    a.x = xr[kb + koff];
    a.y = xr[kb + koff + 1];
    b.x = Wm[(size_t)(kb + koff)     * DOUT + col];
    b.y = Wm[(size_t)(kb + koff + 1) * DOUT + col];
    c = __builtin_amdgcn_wmma_f32_16x16x4_f32(
        /*neg_a=*/false, a, /*neg_b=*/false, b,
        /*c_mod=*/(short)0, c, /*reuse_a=*/false, /*reuse_b=*/false);
  }

  const int mbase = tileM * 16 + ((lane >> 4) << 3);
  const int ncol  = tileN * 16 + (lane & 15);
#pragma unroll
  for (int r = 0; r < 8; ++r) {
    int m = mbase + r;
    if (m < nN) H[(size_t)m * DOUT + ncol] = c[r];
  }
}

// ------------------------------------- out[:,k,:] = b + dinv^2 * H  (init) ---
// Folds the self-loop term and fully overwrites the poisoned output slice.
__global__ void init_out_kernel(const float* __restrict__ H,
                                const float* __restrict__ bias,
                                const float* __restrict__ dinv,
                                float* __restrict__ out,
                                int kslice, int nN)
{
  int idx = blockIdx.x * blockDim.x + threadIdx.x;  // over nN*32 float4 lanes
  int n = idx >> 5, q = idx & 31;
  if (n >= nN) return;
  float dn = dinv[n];
  float w  = dn * dn;
  float4 h  = reinterpret_cast<const float4*>(H + (size_t)n * DOUT)[q];
  float4 bb = reinterpret_cast<const float4*>(bias)[q];
  float4 r;
  r.x = bb.x + w * h.x;  r.y = bb.y + w * h.y;
  r.z = bb.z + w * h.z;  r.w = bb.w + w * h.w;
  reinterpret_cast<float4*>(out + ((size_t)n * K_SLICES + kslice) * DOUT)[q] = r;
}

// --------------------------- out[dst,k,:] += norm * H[src,:]  (edge scatter) ---
// One wave per edge; each lane handles a float4 (32*4 = 128 features).
__global__ __launch_bounds__(256) void scatter_kernel(
    const float* __restrict__ H,
    const int* __restrict__ src, const int* __restrict__ dst,
    const float* __restrict__ dinv,
    float* __restrict__ out, int kslice, int nE)
{
  int e = blockIdx.x * (blockDim.x >> 5) + (threadIdx.x >> 5);
  if (e >= nE) return;
  int lane = threadIdx.x & 31;
  int s = src[e], d = dst[e];
  float nrm = dinv[s] * dinv[d];
  float4 v = reinterpret_cast<const float4*>(H + (size_t)s * DOUT)[lane];
  float* o = out + ((size_t)d * K_SLICES + kslice) * DOUT + lane * 4;
  unsafeAtomicAdd(o + 0, nrm * v.x);
  unsafeAtomicAdd(o + 1, nrm * v.y);
  unsafeAtomicAdd(o + 2, nrm * v.z);
  unsafeAtomicAdd(o + 3, nrm * v.w);
}

// ----------------------------------------------------------------- ReLU -----
__global__ void relu_kernel(float* __restrict__ out, size_t n4) {
  size_t i = (size_t)blockIdx.x * blockDim.x + threadIdx.x;
  if (i < n4) {
    float4* p = reinterpret_cast<float4*>(out) + i;
    float4 v = *p;
    v.x = fmaxf(v.x, 0.f);  v.y = fmaxf(v.y, 0.f);
    v.z = fmaxf(v.z, 0.f);  v.w = fmaxf(v.w, 0.f);
    *p = v;
  }
}

// --------------------------------------------------------------- launcher ---
extern "C" void kernel_launch(void* const* d_in, const int* in_sizes, int n_in,
                              void* d_out, int out_size, void* d_ws, size_t ws_size,
                              hipStream_t stream) {
  const float* x    = (const float*)d_in[0];   // [N, K, DIN]
  const int*   ei   = (const int*)d_in[1];     // [2, E]
  const float* Wm   = (const float*)d_in[2];   // [DIN, DOUT]
  const float* bias = (const float*)d_in[3];   // [DOUT]
  float*       out  = (float*)d_out;           // [N, K, DOUT]

  const int nN = in_sizes[0] / (K_SLICES * DIN);
  const int nE = in_sizes[1] / 2;
  const int* src = ei;
  const int* dst = ei + nE;

  // Workspace: dinv[N] then H[N x DOUT]
  float* dinv = (float*)d_ws;
  size_t off  = ((size_t)nN * sizeof(float) + 255) & ~(size_t)255;
  float* H    = (float*)((char*)d_ws + off);

  // 1) degrees -> dinv
  deg_init_kernel <<<(nN + 255) / 256, 256, 0, stream>>>(dinv, nN);
  deg_count_kernel<<<(nE + 255) / 256, 256, 0, stream>>>(dst, dinv, nE);
  deg_rsqrt_kernel<<<(nN + 255) / 256, 256, 0, stream>>>(dinv, nN);

  // 2) per-slice: WMMA GEMM -> init (bias + self-loop) -> edge scatter
  const int tilesM     = (nN + 15) / 16;
  const int totalWaves = tilesM * (DOUT / 16);
  const int gemmBlocks = (totalWaves + 3) / 4;           // 4 waves / block
  const int initBlocks = (nN * 32 + 255) / 256;
  const int scatBlocks = (nE + 7) / 8;                   // 8 waves / block

  for (int k = 0; k < K_SLICES; ++k) {
    gemm_wmma_kernel<<<gemmBlocks, 128, 0, stream>>>(x, Wm, H, k, nN);
    init_out_kernel <<<initBlocks, 256, 0, stream>>>(H, bias, dinv, out, k, nN);
    scatter_kernel  <<<scatBlocks, 256, 0, stream>>>(H, src, dst, dinv, out, k, nE);
  }

  // 3) ReLU over entire output
  const size_t n4 = (size_t)nN * K_SLICES * DOUT / 4;
  relu_kernel<<<(unsigned)((n4 + 255) / 256), 256, 0, stream>>>(out, n4);
}